// SparseFFN_44341242364339
// MI455X (gfx1250) — compile-verified
//
#include <hip/hip_runtime.h>
#include <hip/hip_bf16.h>

typedef float v2f __attribute__((ext_vector_type(2)));
typedef float v8f __attribute__((ext_vector_type(8)));

#define CHUNKSZ 128
#define NBITS   6
#define KT      32     // K panel depth staged in LDS
#define NBLK    128    // N columns per workgroup
#define XPAD    36     // X row stride (dwords): 16B-aligned rows, conflict-free col reads
#define NW2     144    // packed W row length: 144*2 dw stride => half-waves on disjoint banks

// ---------------- Router: SimHash over chunk means -> (e1, e2) ----------------
__global__ __launch_bounds__(256) void router_kernel(const float* __restrict__ x,
                                                     const float* __restrict__ hp,
                                                     int* __restrict__ eidx, int D) {
    __shared__ float sproj[NBITS];
    const int tid   = threadIdx.x;
    const int chunk = blockIdx.x;
    if (tid < NBITS) sproj[tid] = 0.0f;
    __syncthreads();

    const size_t row0 = (size_t)chunk * CHUNKSZ;
    float p[NBITS] = {0.f, 0.f, 0.f, 0.f, 0.f, 0.f};
    const int ncols = D / 256;                      // 4 columns per thread
    for (int i = 0; i < ncols; ++i) {
        const int d = tid + i * 256;
        const float* col = x + row0 * (size_t)D + d;
        float s = 0.0f;
#pragma unroll 8
        for (int r = 0; r < CHUNKSZ; ++r) s += col[(size_t)r * D];
#pragma unroll
        for (int j = 0; j < NBITS; ++j) p[j] += s * hp[d * NBITS + j];
    }
#pragma unroll
    for (int j = 0; j < NBITS; ++j) atomicAdd(&sproj[j], p[j]);   // ds_add_f32
    __syncthreads();

    if (tid == 0) {
        float proj[NBITS];
        int e1 = 0;
#pragma unroll
        for (int j = 0; j < NBITS; ++j) {
            proj[j] = sproj[j] * (1.0f / CHUNKSZ);
            e1 |= (proj[j] > 0.0f) ? (1 << j) : 0;
        }
        int weakest = 0;
        float best = fabsf(proj[0]);
#pragma unroll
        for (int j = 1; j < NBITS; ++j) {
            const float a = fabsf(proj[j]);
            if (a < best) { best = a; weakest = j; }
        }
        eidx[chunk * 2 + 0] = e1;
        eidx[chunk * 2 + 1] = e1 ^ (1 << weakest);
    }
}

// -------- Chunk GEMM: out = X[128xD] @ 0.5*(W[e1]+W[e2]) via f32 WMMA --------
// Software-pipelined: next panel's global loads are in flight during WMMA.
__global__ __launch_bounds__(256) void moe_gemm_kernel(const float* __restrict__ x,
                                                       const float* __restrict__ w,
                                                       const int* __restrict__ eidx,
                                                       float* __restrict__ out, int D) {
    __shared__ float Xs[CHUNKSZ][XPAD];     // X panel 128 x 32 (padded)      18.4 KB
    __shared__ float Wp[KT / 2][NW2][2];    // combined W, K-pair packed      18.4 KB

    const int tid   = threadIdx.x;
    const int lane  = tid & 31;
    const int wave  = tid >> 5;                 // 0..7 -> M rows wave*16..+15
    const int chunk = blockIdx.y;
    const int n0    = blockIdx.x * NBLK;
    const int m0    = wave * 16;

    const int e1 = eidx[chunk * 2 + 0];
    const int e2 = eidx[chunk * 2 + 1];
    const size_t wb1   = (size_t)e1 * D * D;
    const size_t wb2   = (size_t)e2 * D * D;
    const size_t xrow0 = (size_t)chunk * CHUNKSZ;

    v8f acc[8] = {};                            // wave covers a 16 x 128 output slab

    // ---- register staging for the software pipeline ----
    float4 xr[4], wa[4], wb[4];

    // Prologue: load panel 0 into registers
#pragma unroll
    for (int i = 0; i < 4; ++i) {
        const int idx = tid + i * 256;
        xr[i] = *(const float4*)(x + (xrow0 + (idx >> 3)) * (size_t)D + ((idx & 7) * 4));
        const size_t off = (size_t)(idx >> 5) * D + n0 + (idx & 31) * 4;
        wa[i] = *(const float4*)(w + wb1 + off);
        wb[i] = *(const float4*)(w + wb2 + off);
    }

    const int nPanels = D / KT;
    for (int p = 0; p < nPanels; ++p) {
        // ---- drain staged registers into LDS ----
#pragma unroll
        for (int i = 0; i < 4; ++i) {
            const int idx = tid + i * 256;
            {   // X: 128 rows x 32 cols, b128 stores
                const int r = idx >> 3;
                const int c = (idx & 7) * 4;
                *(float4*)&Xs[r][c] = xr[i];
            }
            {   // W: combine 0.5*(W1+W2), pair-packed so B-frags are one b64
                const int r = idx >> 5;         // k row 0..31
                const int c = (idx & 31) * 4;   // n col
                const int rh = r >> 1, rl = r & 1;
                Wp[rh][c + 0][rl] = 0.5f * (wa[i].x + wb[i].x);
                Wp[rh][c + 1][rl] = 0.5f * (wa[i].y + wb[i].y);
                Wp[rh][c + 2][rl] = 0.5f * (wa[i].z + wb[i].z);
                Wp[rh][c + 3][rl] = 0.5f * (wa[i].w + wb[i].w);
            }
        }
        __syncthreads();

        // ---- issue next panel's global loads (overlap with WMMA below) ----
        if (p + 1 < nPanels) {
            const int k0n = (p + 1) * KT;
#pragma unroll
            for (int i = 0; i < 4; ++i) {
                const int idx = tid + i * 256;
                xr[i] = *(const float4*)(x + (xrow0 + (idx >> 3)) * (size_t)D +
                                         (k0n + (idx & 7) * 4));
                const size_t off = (size_t)(k0n + (idx >> 5)) * D + n0 + (idx & 31) * 4;
                wa[i] = *(const float4*)(w + wb1 + off);
                wb[i] = *(const float4*)(w + wb2 + off);
            }
        }

        // ---- WMMA compute on current panel ----
        // A 16x4 layout: lanes 0-15 carry K=0,1 ; lanes 16-31 carry K=2,3
        const int mrow = m0 + (lane & 15);
        const int ko   = (lane >> 4) * 2;
#pragma unroll
        for (int kk = 0; kk < KT; kk += 4) {
            const v2f afrag = *(const v2f*)&Xs[mrow][kk + ko];      // ds_load_b64
#pragma unroll
            for (int t = 0; t < 8; ++t) {
                const int ncol = t * 16 + (lane & 15);
                const v2f bfrag = *(const v2f*)&Wp[(kk + ko) >> 1][ncol][0]; // ds_load_b64
                acc[t] = __builtin_amdgcn_wmma_f32_16x16x4_f32(
                    /*neg_a=*/false, afrag, /*neg_b=*/false, bfrag,
                    /*c_mod=*/(short)0, acc[t], /*reuse_a=*/false, /*reuse_b=*/false);
            }
        }
        __syncthreads();
    }

    // C/D layout: VGPR r -> M = m0 + r (lanes 0-15) / m0 + r + 8 (lanes 16-31)
    const int rbase = m0 + ((lane >> 4) * 8);
#pragma unroll
    for (int t = 0; t < 8; ++t) {
        const int col = n0 + t * 16 + (lane & 15);
#pragma unroll
        for (int r = 0; r < 8; ++r) {
            out[(xrow0 + rbase + r) * (size_t)D + col] = acc[t][r];
        }
    }
}

extern "C" void kernel_launch(void* const* d_in, const int* in_sizes, int n_in,
                              void* d_out, int out_size, void* d_ws, size_t ws_size,
                              hipStream_t stream) {
    const float* x  = (const float*)d_in[0];   // [B, L, D] f32
    const float* w  = (const float*)d_in[1];   // [E, D, D] f32
    const float* hp = (const float*)d_in[2];   // [D, NBITS] f32
    float* out = (float*)d_out;

    const int D       = in_sizes[2] / NBITS;       // 1024
    const int rows    = in_sizes[0] / D;           // B*L = 16384
    const int nChunks = rows / CHUNKSZ;            // 128
    int* eidx = (int*)d_ws;                        // [nChunks, 2] expert indices

    router_kernel<<<nChunks, 256, 0, stream>>>(x, hp, eidx, D);

    dim3 grid(D / NBLK, nChunks);                  // (8, 128)
    moe_gemm_kernel<<<grid, 256, 0, stream>>>(x, w, eidx, out, D);
}